// Model_47390669144550
// MI455X (gfx1250) — compile-verified
//
#include <hip/hip_runtime.h>
#include <float.h>

#ifndef __has_builtin
#define __has_builtin(x) 0
#endif

// Problem constants (from reference setup_inputs)
#define B_    128
#define NP_   16384
#define NC_   4096
#define M_    (NP_ + NC_)        // 20480 points per batch
#define S_    2048               // merge_points
#define TPB   1024               // 32 wave32 waves
#define PTS   (M_ / TPB)         // 20 points per thread
#define WAVES (TPB / 32)         // 32

// Per-thread LDS block: x[20] | y[20] | z[20] with 1 pad DWORD after each
// group -> 63 DWORDs. 63 is coprime with 64 banks => conflict-free, and all
// hot-loop accesses are immediate DS offsets from a single base VGPR.
#define TSTRIDE 63
#define YOFF    21
#define ZOFF    42

// Dynamic LDS: pts[TPB*63] | redv[WAVES] | redi[WAVES] | winner[1]
#define SMEM_FLOATS (TPB * TSTRIDE + WAVES)
#define SMEM_BYTES  (SMEM_FLOATS * sizeof(float) + (WAVES + 1) * sizeof(int))

// ---- CDNA5 async global->LDS staging (guarded; falls back to plain loads) ----
#if __has_builtin(__builtin_amdgcn_global_load_async_to_lds_b32)
#define FPS_USE_ASYNC 1
#endif

__device__ __forceinline__ void fps_wait_async() {
#if defined(FPS_USE_ASYNC)
#if __has_builtin(__builtin_amdgcn_s_wait_asynccnt)
  __builtin_amdgcn_s_wait_asynccnt(0);
#else
  asm volatile("s_wait_asynccnt 0" ::: "memory");
#endif
#endif
}

// Wave32 argmax reduction; ties resolved toward the smaller index (jnp.argmax).
__device__ __forceinline__ void wave_argmax(float& v, int& i) {
#pragma unroll
  for (int off = 16; off > 0; off >>= 1) {
    float ov = __shfl_down(v, off, 32);
    int   oi = __shfl_down(i, off, 32);
    if (ov > v || (ov == v && oi < i)) { v = ov; i = oi; }
  }
}

__global__ __launch_bounds__(TPB, 1) void fps_gather_kernel(
    const float* __restrict__ partial,   // [B, NP, 3]
    const float* __restrict__ coarse,    // [B, 3, NC]
    float* __restrict__ out) {           // [B, 3, S]
  extern __shared__ float smem[];
  float* redv   = smem + TPB * TSTRIDE;
  int*   redi   = (int*)(redv + WAVES);
  int*   winner = redi + WAVES;

  const int tid  = threadIdx.x;
  const int b    = blockIdx.x;
  const int base = tid * TSTRIDE;             // this thread's LDS block
  float* sp = smem + base;

  const float* pb = partial + (size_t)b * NP_ * 3;
  const float* cb = coarse  + (size_t)b * 3 * NC_;

  // ---------- Stage points: thread tid owns m in [20*tid, 20*tid+20) ----------
#pragma unroll
  for (int k = 0; k < PTS; ++k) {
    const int m = tid * PTS + k;
    const float* srcx;
    const float* srcy;
    const float* srcz;
    if (m < NP_) {                 // from partial[b, m, :]
      srcx = pb + 3 * m;
      srcy = srcx + 1;
      srcz = srcx + 2;
    } else {                       // from coarse[b, :, m-NP]
      const int j = m - NP_;
      srcx = cb + j;
      srcy = cb + NC_ + j;
      srcz = cb + 2 * NC_ + j;
    }
#if defined(FPS_USE_ASYNC)
    __builtin_amdgcn_global_load_async_to_lds_b32((int*)srcx, (int*)&sp[k], 0, 0);
    __builtin_amdgcn_global_load_async_to_lds_b32((int*)srcy, (int*)&sp[YOFF + k], 0, 0);
    __builtin_amdgcn_global_load_async_to_lds_b32((int*)srcz, (int*)&sp[ZOFF + k], 0, 0);
#else
    sp[k]        = *srcx;
    sp[YOFF + k] = *srcy;
    sp[ZOFF + k] = *srcz;
#endif
  }
  fps_wait_async();
  __syncthreads();

  // Per-thread min-distance state in VGPRs
  float md[PTS];
#pragma unroll
  for (int k = 0; k < PTS; ++k) md[k] = FLT_MAX;

  int last = 0;  // first selected index is 0 (matches reference scan semantics)
  for (int s = 0; s < S_; ++s) {
    // Locate selected point's LDS slot: m = 20*t + kq  ->  dword t*63 + kq
    const int t  = last / PTS;
    const int kq = last - t * PTS;
    const int qi = t * TSTRIDE + kq;

    // Fused gather: emit the currently-selected point into out[b, :, s]
    if (tid == 0)      out[((size_t)b * 3 + 0) * S_ + s] = smem[qi];
    else if (tid == 1) out[((size_t)b * 3 + 1) * S_ + s] = smem[qi + YOFF];
    else if (tid == 2) out[((size_t)b * 3 + 2) * S_ + s] = smem[qi + ZOFF];

    const float qx = smem[qi];          // LDS broadcast reads
    const float qy = smem[qi + YOFF];
    const float qz = smem[qi + ZOFF];

    float bv = -1.0f;
    int   bk = 0;
#pragma unroll
    for (int k = 0; k < PTS; ++k) {     // immediate-offset, conflict-free DS loads
      const float dx = sp[k]        - qx;
      const float dy = sp[YOFF + k] - qy;
      const float dz = sp[ZOFF + k] - qz;
      const float d  = dx * dx + dy * dy + dz * dz;
      const float nm = fminf(md[k], d);
      md[k] = nm;
      if (nm > bv) { bv = nm; bk = k; }  // strict '>' keeps first max (k ascends)
    }
    int bi = tid * PTS + bk;

    // Intra-wave argmax (wave32 shuffle tree)
    wave_argmax(bv, bi);
    if ((tid & 31) == 0) { redv[tid >> 5] = bv; redi[tid >> 5] = bi; }
    __syncthreads();

    // Cross-wave argmax: wave 0 reduces the 32 per-wave winners
    if (tid < WAVES) {
      float v = redv[tid];
      int   i = redi[tid];
      wave_argmax(v, i);
      if (tid == 0) *winner = i;
    }
    __syncthreads();
    last = __builtin_amdgcn_readfirstlane(*winner);  // provably uniform
  }
}

extern "C" void kernel_launch(void* const* d_in, const int* in_sizes, int n_in,
                              void* d_out, int out_size, void* d_ws, size_t ws_size,
                              hipStream_t stream) {
  const float* partial = (const float*)d_in[0];  // [B, NP, 3]
  const float* coarse  = (const float*)d_in[1];  // [B, 3, NC]
  float* out = (float*)d_out;                    // [B, 3, S]
  (void)in_sizes; (void)n_in; (void)out_size; (void)d_ws; (void)ws_size;

  // Allow >64KB dynamic LDS (idempotent; not a stream op, graph-capture safe)
  (void)hipFuncSetAttribute(reinterpret_cast<const void*>(fps_gather_kernel),
                            hipFuncAttributeMaxDynamicSharedMemorySize,
                            (int)SMEM_BYTES);

  fps_gather_kernel<<<B_, TPB, SMEM_BYTES, stream>>>(partial, coarse, out);
}